// GNN_20770461844169
// MI455X (gfx1250) — compile-verified
//
#include <hip/hip_runtime.h>
#include <hip/hip_bf16.h>

typedef __attribute__((ext_vector_type(2))) float v2f;
typedef __attribute__((ext_vector_type(8))) float v8f;

#define HFEAT 64

// ---------------- degree kernels ----------------
__global__ void k_fill1(float* d, int n) {
    int t = blockIdx.x * blockDim.x + threadIdx.x;
    if (t < n) d[t] = 1.0f;  // self loop contributes 1 to every node's in-degree
}

__global__ void k_deg(const int* __restrict__ dst, float* __restrict__ deg, int e) {
    int t = blockIdx.x * blockDim.x + threadIdx.x;
    if (t < e) unsafeAtomicAdd(&deg[dst[t]], 1.0f);
}

__global__ void k_rsqrt(float* d, int n) {
    int t = blockIdx.x * blockDim.x + threadIdx.x;
    if (t < n) d[t] = rsqrtf(d[t]);  // deg >= 1 always (self loop)
}

// ---------------- WMMA f32 GEMM: C[M x NC] = A[M x K] @ B[K x NC] ----------------
// One wave computes one 16x16 output tile with v_wmma_f32_16x16x4_f32.
// A 16x4 layout: lane L holds row M=L%16, K = (L/16)*2 + {0,1} in the v2f pair.
// B  4x16 layout: lane L holds col N=L%16, K = (L/16)*2 + {0,1}.
// C 16x16 layout: acc[i] -> row = i + 8*(L/16), col = L%16.
template <int K, int NC>
__global__ void k_gemm_wmma(const float* __restrict__ A, const float* __restrict__ B,
                            float* __restrict__ C, int M) {
    const int NT = NC / 16;
    int wave = (int)((blockIdx.x * blockDim.x + threadIdx.x) >> 5);
    int lane = threadIdx.x & 31;
    int totalTiles = (M / 16) * NT;
    if (wave >= totalTiles) return;  // wave-uniform: EXEC stays all-1s for WMMA

    int m0 = (wave / NT) * 16;
    int n0 = (wave % NT) * 16;
    int row  = m0 + (lane & 15);
    int col  = n0 + (lane & 15);
    int koff = (lane >> 4) * 2;

    v8f acc = {};
    const float* arow = A + (size_t)row * K + koff;
#pragma unroll 4
    for (int k = 0; k < K; k += 4) {
        v2f a = *(const v2f*)(arow + k);
        v2f b;
        b.x = B[(size_t)(k + koff)     * NC + col];
        b.y = B[(size_t)(k + koff + 1) * NC + col];
        acc = __builtin_amdgcn_wmma_f32_16x16x4_f32(false, a, false, b,
                                                    (short)0, acc, false, false);
    }

    int rbase = m0 + ((lane >> 4) * 8);
#pragma unroll
    for (int i = 0; i < 8; ++i)
        C[(size_t)(rbase + i) * NC + col] = acc[i];
}

// ---------------- aggregation ----------------
// Initialize accumulator with the self-loop term: out[i,:] = dinv[i]^2 * xw[i,:]
__global__ void k_self_init(const float* __restrict__ xw, const float* __restrict__ dinv,
                            float* __restrict__ out, int total) {
    int t = blockIdx.x * blockDim.x + threadIdx.x;
    if (t >= total) return;
    int node = t >> 6;  // HFEAT == 64
    float di = dinv[node];
    out[t] = di * di * xw[t];
}

// Edge scatter-add: 16 lanes per edge, float4 per lane (64 feats/edge).
__global__ void k_edge_scatter(const int* __restrict__ src, const int* __restrict__ dst,
                               const float* __restrict__ dinv,
                               const float* __restrict__ xw, float* __restrict__ out, int e) {
    int t = blockIdx.x * blockDim.x + threadIdx.x;
    int edge = t >> 4;
    if (edge >= e) return;
    int f4 = (t & 15) * 4;
    int s = src[edge];
    int d = dst[edge];
    float c = dinv[s] * dinv[d];
    const float4 v = *(const float4*)(xw + (size_t)s * HFEAT + f4);
    float* o = out + (size_t)d * HFEAT + f4;
    unsafeAtomicAdd(o + 0, c * v.x);
    unsafeAtomicAdd(o + 1, c * v.y);
    unsafeAtomicAdd(o + 2, c * v.z);
    unsafeAtomicAdd(o + 3, c * v.w);
}

__global__ void k_bias_relu(float* __restrict__ h, const float* __restrict__ b, int total) {
    int t = blockIdx.x * blockDim.x + threadIdx.x;
    if (t >= total) return;
    float v = h[t] + b[t & (HFEAT - 1)];
    h[t] = v > 0.0f ? v : 0.0f;
}

// ---------------- fused head: out[i] = relu(g[i,:] + fc1_b) . fco_W + fco_b ----------------
// One wave32 per node; lane j owns fc1 output channel j.
__global__ void k_fc_head(const float* __restrict__ g, const float* __restrict__ fc1_b,
                          const float* __restrict__ fcoW, const float* __restrict__ fcob,
                          float* __restrict__ out, int n) {
    int wave = (int)((blockIdx.x * blockDim.x + threadIdx.x) >> 5);
    int lane = threadIdx.x & 31;
    if (wave >= n) return;
    float v = g[(size_t)wave * 32 + lane] + fc1_b[lane];
    v = v > 0.0f ? v : 0.0f;
    float p = v * fcoW[lane];
#pragma unroll
    for (int off = 16; off > 0; off >>= 1) p += __shfl_xor(p, off, 32);
    if (lane == 0) out[wave] = p + fcob[0];
}

// ---------------- launch ----------------
extern "C" void kernel_launch(void* const* d_in, const int* in_sizes, int n_in,
                              void* d_out, int out_size, void* d_ws, size_t ws_size,
                              hipStream_t stream) {
    const float* x     = (const float*)d_in[0];
    const int*   ei    = (const int*)d_in[1];   // [2, E] int32
    const float* W1    = (const float*)d_in[2];
    const float* b1    = (const float*)d_in[3];
    const float* W2    = (const float*)d_in[4];
    const float* b2    = (const float*)d_in[5];
    const float* fc1W  = (const float*)d_in[6];
    const float* fc1b  = (const float*)d_in[7];
    const float* fcoW  = (const float*)d_in[8];
    const float* fcob  = (const float*)d_in[9];
    float* out = (float*)d_out;

    const int N = in_sizes[0] / 128;   // 100000
    const int E = in_sizes[1] / 2;     // 1600000
    const int* src = ei;
    const int* dst = ei + E;

    float* dinv = (float*)d_ws;
    float* bufA = dinv + N;                   // GEMM outputs [N x 64]
    float* bufB = bufA + (size_t)N * HFEAT;   // aggregation accumulator [N x 64]

    const int B256 = 256;
    auto blocks = [](long long work, int bs) { return (int)((work + bs - 1) / bs); };

    // degrees -> dinv
    k_fill1<<<blocks(N, B256), B256, 0, stream>>>(dinv, N);
    k_deg<<<blocks(E, B256), B256, 0, stream>>>(dst, dinv, E);
    k_rsqrt<<<blocks(N, B256), B256, 0, stream>>>(dinv, N);

    // ---- layer 1: xw = x @ W1 ; agg ; +b1 ; relu ----
    {
        long long waves = (long long)(N / 16) * (HFEAT / 16);
        k_gemm_wmma<128, HFEAT><<<blocks(waves * 32, B256), B256, 0, stream>>>(x, W1, bufA, N);
    }
    k_self_init<<<blocks((long long)N * HFEAT, B256), B256, 0, stream>>>(bufA, dinv, bufB, N * HFEAT);
    k_edge_scatter<<<blocks((long long)E * 16, B256), B256, 0, stream>>>(src, dst, dinv, bufA, bufB, E);
    k_bias_relu<<<blocks((long long)N * HFEAT, B256), B256, 0, stream>>>(bufB, b1, N * HFEAT);

    // ---- layer 2: xw = h @ W2 ; agg ; +b2 ; relu ----
    {
        long long waves = (long long)(N / 16) * (HFEAT / 16);
        k_gemm_wmma<HFEAT, HFEAT><<<blocks(waves * 32, B256), B256, 0, stream>>>(bufB, W2, bufA, N);
    }
    k_self_init<<<blocks((long long)N * HFEAT, B256), B256, 0, stream>>>(bufA, dinv, bufB, N * HFEAT);
    k_edge_scatter<<<blocks((long long)E * 16, B256), B256, 0, stream>>>(src, dst, dinv, bufA, bufB, E);
    k_bias_relu<<<blocks((long long)N * HFEAT, B256), B256, 0, stream>>>(bufB, b2, N * HFEAT);

    // ---- head: g = h @ fc1_W ; out = relu(g + fc1_b) . fco_W + fco_b ----
    {
        long long waves = (long long)(N / 16) * (32 / 16);
        k_gemm_wmma<HFEAT, 32><<<blocks(waves * 32, B256), B256, 0, stream>>>(bufB, fc1W, bufA, N);
    }
    k_fc_head<<<blocks((long long)N * 32, B256), B256, 0, stream>>>(bufA, fc1b, fcoW, fcob, out, N);
}